// LaneATTHead_45028437131817
// MI455X (gfx1250) — compile-verified
//
#include <hip/hip_runtime.h>
#include <hip/hip_bf16.h>

// ---------------------------------------------------------------------------
// LaneATT head for MI455X (gfx1250).
//  - fp32 matrix math on V_WMMA_F32_16X16X4_F32 (only f32 WMMA on CDNA5)
//  - GEMM tile feeds on the gfx1250 async-to-LDS path (ASYNCcnt + double buffer)
// ---------------------------------------------------------------------------

typedef __attribute__((ext_vector_type(2))) float v2f;
typedef __attribute__((ext_vector_type(8))) float v8f;
typedef int v4i_vs __attribute__((vector_size(16)));   // matches builtin proto

#define BB   16
#define CIN  64
#define HH   160
#define WW   192
#define HWP  (HH*WW)          // 30720
#define NN   1000
#define NM1  999
#define AFC  16
#define SS   96
#define NCC  2
#define DD   (AFC*HH)         // 2560
#define DD2  (2*DD)           // 5120
#define OUTH (NCC + 3*SS)     // 290
#define PROP (NCC + 3 + 3*SS) // 293

// LDS row pitch (floats): 36 -> 144B rows = 16B aligned for b128 async stores,
// and 36*r mod 64 covers all 16 bank groups -> conflict-free WMMA reads.
#define PAD  36

#if defined(__gfx1250__) && __has_builtin(__builtin_amdgcn_global_load_async_to_lds_b128)
#define USE_ASYNC 1
#endif

#ifdef USE_ASYNC
#define ASYNC_B128(gptr, lptr)                                                  \
    __builtin_amdgcn_global_load_async_to_lds_b128(                             \
        (__attribute__((address_space(1))) v4i_vs*)(void*)(gptr),               \
        (__attribute__((address_space(3))) v4i_vs*)(void*)(lptr), 0, 0)
#if __has_builtin(__builtin_amdgcn_s_wait_asynccnt)
#define WAIT_ASYNC(n) __builtin_amdgcn_s_wait_asynccnt(n)
#else
#define WAIT_ASYNC(n) asm volatile("s_wait_asynccnt %0" ::"n"(n) : "memory")
#endif
#endif

// ---------------------------------------------------------------------------
// WMMA inner chunk helpers (consume one 64x32 A slab + 32-wide W/B slab)
// ---------------------------------------------------------------------------
__device__ __forceinline__ void wmma_chunk_nt(const float* __restrict__ As,
                                              const float* __restrict__ Ws,
                                              int wm, int wn, int l16, int hi, v8f& c)
{
#pragma unroll
    for (int kk = 0; kk < 32; kk += 4) {
        v2f a, b;
        const float* ap = &As[(wm*16 + l16)*PAD + kk + 2*hi];
        a.x = ap[0]; a.y = ap[1];
        const float* bp = &Ws[(wn*16 + l16)*PAD + kk + hi];   // Ws[n][k]
        b.x = bp[0]; b.y = bp[2];
        c = __builtin_amdgcn_wmma_f32_16x16x4_f32(false, a, false, b,
                                                  (short)0, c, false, false);
    }
}

__device__ __forceinline__ void wmma_chunk_nn(const float* __restrict__ As,
                                              const float* __restrict__ Bs,
                                              int wm, int wn, int l16, int hi, v8f& c)
{
#pragma unroll
    for (int kk = 0; kk < 32; kk += 4) {
        v2f a, b;
        const float* ap = &As[(wm*16 + l16)*PAD + kk + 2*hi];
        a.x = ap[0]; a.y = ap[1];
        const float* bp = &Bs[(kk + hi)*PAD + wn*16 + l16];   // Bs[k][n]
        b.x = bp[0]; b.y = bp[2*PAD];
        c = __builtin_amdgcn_wmma_f32_16x16x4_f32(false, a, false, b,
                                                  (short)0, c, false, false);
    }
}

// ---------------------------------------------------------------------------
// Kernel 1: 1x1 conv as WMMA GEMM: feats(16 x pix) = conv_w(16x64) * X(64 x pix)
// ---------------------------------------------------------------------------
__global__ void conv1x1_wmma(const float* __restrict__ bf,
                             const float* __restrict__ cw,
                             const float* __restrict__ cb,
                             float* __restrict__ feats)
{
    const int wave = threadIdx.x >> 5;
    const int lane = threadIdx.x & 31;
    const int l16  = lane & 15;
    const int hi   = lane >> 4;
    const int tile = blockIdx.x * 8 + wave;
    const int pix0 = tile * 16;
    const int bidx = pix0 / HWP;
    const int off0 = pix0 - bidx * HWP;

    v8f c;
#pragma unroll
    for (int r = 0; r < 8; ++r) c[r] = cb[r + 8*hi];

    const float* xb = bf + (size_t)bidx * CIN * HWP + off0;
#pragma unroll
    for (int kk = 0; kk < CIN; kk += 4) {
        v2f a, b;
        a.x = cw[l16*CIN + kk + 2*hi];
        a.y = cw[l16*CIN + kk + 2*hi + 1];
        b.x = xb[(size_t)(kk + hi)     * HWP + l16];
        b.y = xb[(size_t)(kk + 2 + hi) * HWP + l16];
        c = __builtin_amdgcn_wmma_f32_16x16x4_f32(false, a, false, b,
                                                  (short)0, c, false, false);
    }

    float* fo = feats + (size_t)bidx * AFC * HWP + off0 + l16;
#pragma unroll
    for (int r = 0; r < 8; ++r)
        fo[(size_t)(r + 8*hi) * HWP] = c[r];
}

// ---------------------------------------------------------------------------
// Kernel 2: gather ROI columns -> baf[b][n][a*H + h]
// ---------------------------------------------------------------------------
__global__ void gather_baf(const float* __restrict__ feats,
                           const int* __restrict__ cut_xs,
                           const unsigned char* __restrict__ inval,
                           float* __restrict__ baf)
{
    int idx = blockIdx.x * blockDim.x + threadIdx.x;
    if (idx >= BB*NN*HH) return;
    int h = idx % HH;
    int t = idx / HH;
    int n = t % NN;
    int b = t / NN;
    int x = cut_xs[n*HH + h];
    bool bad = inval[n*HH + h] != 0;
    const float* fp = feats + (size_t)b * AFC * HWP + (size_t)h * WW + x;
    float* op = baf + ((size_t)(b*NN + n)) * DD + h;
#pragma unroll
    for (int a = 0; a < AFC; ++a)
        op[a*HH] = bad ? 0.0f : fp[(size_t)a * HWP];
}

// ---------------------------------------------------------------------------
// Kernel 3: NT GEMM  C[M x Nw] = A[M x K] * W[Nw x K]^T + bias
// 64x32 block tile, 8 waves; async-to-LDS double-buffered K pipeline.
// Requires K % 32 == 0, M % 64 == 0 (true for all uses).
// ---------------------------------------------------------------------------
__global__ void gemm_nt_wmma(const float* __restrict__ A,
                             const float* __restrict__ Wt,
                             const float* __restrict__ bias,
                             float* __restrict__ C,
                             int M, int Nw, int K)
{
    __shared__ float As[2][64*PAD];
    __shared__ float Ws[2][32*PAD];
    const int t    = threadIdx.x;
    const int m0   = blockIdx.y * 64;
    const int n0   = blockIdx.x * 32;
    const int wave = t >> 5, lane = t & 31, l16 = lane & 15, hi = lane >> 4;
    const int wm   = wave >> 1, wn = wave & 1;

    v8f c = {0.f,0.f,0.f,0.f,0.f,0.f,0.f,0.f};

#ifdef USE_ASYNC
    // per-thread async segment map (b128 = 4 floats)
    const int ra0 = t >> 3,         ga0 = t & 7;          // A seg t
    const int ra1 = (t + 256) >> 3, ga1 = (t + 256) & 7;  // A seg t+256
    const int rw  = t >> 3,         gw  = t & 7;          // W seg t
    int nrow = n0 + rw; if (nrow >= Nw) nrow = Nw - 1;    // clamp: junk cols discarded

    const int nch = K / 32;
    {   // prologue: chunk 0 -> buffer 0
        ASYNC_B128(A  + (size_t)(m0 + ra0)*K + ga0*4, &As[0][ra0*PAD + ga0*4]);
        ASYNC_B128(A  + (size_t)(m0 + ra1)*K + ga1*4, &As[0][ra1*PAD + ga1*4]);
        ASYNC_B128(Wt + (size_t)nrow     *K + gw *4, &Ws[0][rw *PAD + gw *4]);
    }
    for (int i = 0; i < nch; ++i) {
        const int cur = i & 1;
        if (i + 1 < nch) {
            const int k1 = (i + 1) * 32;
            const int nb = (i + 1) & 1;
            ASYNC_B128(A  + (size_t)(m0 + ra0)*K + k1 + ga0*4, &As[nb][ra0*PAD + ga0*4]);
            ASYNC_B128(A  + (size_t)(m0 + ra1)*K + k1 + ga1*4, &As[nb][ra1*PAD + ga1*4]);
            ASYNC_B128(Wt + (size_t)nrow     *K + k1 + gw *4, &Ws[nb][rw *PAD + gw *4]);
            WAIT_ASYNC(3);
        } else {
            WAIT_ASYNC(0);
        }
        __syncthreads();
        wmma_chunk_nt(As[cur], Ws[cur], wm, wn, l16, hi, c);
        __syncthreads();
    }
#else
    for (int k0 = 0; k0 < K; k0 += 32) {
#pragma unroll
        for (int i = 0; i < 8; ++i) {
            int idx = t + i*256;
            int r = idx >> 5, kk = idx & 31;
            As[0][r*PAD + kk] = A[(size_t)(m0 + r) * K + k0 + kk];
        }
#pragma unroll
        for (int i = 0; i < 4; ++i) {
            int idx = t + i*256;
            int r = idx >> 5, kk = idx & 31;
            int n = n0 + r;
            Ws[0][r*PAD + kk] = (n < Nw) ? Wt[(size_t)n * K + k0 + kk] : 0.0f;
        }
        __syncthreads();
        wmma_chunk_nt(As[0], Ws[0], wm, wn, l16, hi, c);
        __syncthreads();
    }
#endif

    int n = n0 + wn*16 + l16;
    if (n < Nw) {
        float bv = bias ? bias[n] : 0.0f;
#pragma unroll
        for (int r = 0; r < 8; ++r) {
            int m = m0 + wm*16 + r + 8*hi;
            C[(size_t)m * Nw + n] = c[r] + bv;
        }
    }
}

// ---------------------------------------------------------------------------
// Kernel 4: row softmax over 999 columns, in place.
// ---------------------------------------------------------------------------
__global__ void softmax_rows(float* __restrict__ sc)
{
    __shared__ float red[256];
    const int row = blockIdx.x;
    float* p = sc + (size_t)row * NM1;
    const int t = threadIdx.x;

    float m = -1e30f;
    for (int j = t; j < NM1; j += 256) m = fmaxf(m, p[j]);
    red[t] = m; __syncthreads();
    for (int s = 128; s > 0; s >>= 1) {
        if (t < s) red[t] = fmaxf(red[t], red[t+s]);
        __syncthreads();
    }
    m = red[0];
    __syncthreads();

    float sum = 0.0f;
    for (int j = t; j < NM1; j += 256) {
        float e = __expf(p[j] - m);
        p[j] = e;
        sum += e;
    }
    red[t] = sum; __syncthreads();
    for (int s = 128; s > 0; s >>= 1) {
        if (t < s) red[t] += red[t+s];
        __syncthreads();
    }
    float inv = 1.0f / red[0];
    __syncthreads();
    for (int j = t; j < NM1; j += 256) p[j] *= inv;
}

// ---------------------------------------------------------------------------
// Kernel 5: expand attention (B,N,N-1) -> attn_mat (B,N,N), zero diagonal.
// ---------------------------------------------------------------------------
__global__ void build_attn(const float* __restrict__ att, float* __restrict__ out)
{
    size_t idx = (size_t)blockIdx.x * blockDim.x + threadIdx.x;
    if (idx >= (size_t)BB*NN*NN) return;
    int j = (int)(idx % NN);
    size_t tt = idx / NN;
    int i = (int)(tt % NN);
    int b = (int)(tt / NN);
    float v = 0.0f;
    if (i != j) {
        int src = j - (j > i ? 1 : 0);
        v = att[((size_t)(b*NN + i)) * NM1 + src];
    }
    out[idx] = v;
}

// ---------------------------------------------------------------------------
// Kernel 6: batched NN GEMM: attf[b](1000x2560) = attn_mat[b](1000x1000)*baf[b]
// 31 async full chunks (k<992) + 1 guarded synchronous tail chunk (k 992..999).
// ---------------------------------------------------------------------------
__global__ void gemm_nn_wmma(const float* __restrict__ Afull,
                             const float* __restrict__ Bfull,
                             float* __restrict__ Cfull)
{
    __shared__ float As[2][64*PAD];
    __shared__ float Bs[2][32*PAD];
    const int b  = blockIdx.z;
    const float* A  = Afull + (size_t)b * NN * NN;
    const float* Bm = Bfull + (size_t)b * NN * DD;
    float* C        = Cfull + (size_t)b * NN * DD;

    const int t    = threadIdx.x;
    const int m0   = blockIdx.y * 64;                  // up to 1024, guarded
    const int n0   = blockIdx.x * 32;
    const int wave = t >> 5, lane = t & 31, l16 = lane & 15, hi = lane >> 4;
    const int wm   = wave >> 1, wn = wave & 1;

    v8f c = {0.f,0.f,0.f,0.f,0.f,0.f,0.f,0.f};

#ifdef USE_ASYNC
    const int ra0 = t >> 3,         ga0 = t & 7;
    const int ra1 = (t + 256) >> 3, ga1 = (t + 256) & 7;
    const int rb  = t >> 3,         gb  = t & 7;       // Bs: row = k, col seg = n
    int mr0 = m0 + ra0; if (mr0 >= NN) mr0 = NN - 1;   // clamp: junk rows discarded
    int mr1 = m0 + ra1; if (mr1 >= NN) mr1 = NN - 1;

    const int nch = 31;                                // k = 0 .. 991 (all < 1000)
    {
        ASYNC_B128(A  + (size_t)mr0*NN + ga0*4,        &As[0][ra0*PAD + ga0*4]);
        ASYNC_B128(A  + (size_t)mr1*NN + ga1*4,        &As[0][ra1*PAD + ga1*4]);
        ASYNC_B128(Bm + (size_t)rb *DD + n0 + gb*4,    &Bs[0][rb *PAD + gb *4]);
    }
    for (int i = 0; i < nch; ++i) {
        const int cur = i & 1;
        if (i + 1 < nch) {
            const int k1 = (i + 1) * 32;
            const int nb = (i + 1) & 1;
            ASYNC_B128(A  + (size_t)mr0*NN + k1 + ga0*4,          &As[nb][ra0*PAD + ga0*4]);
            ASYNC_B128(A  + (size_t)mr1*NN + k1 + ga1*4,          &As[nb][ra1*PAD + ga1*4]);
            ASYNC_B128(Bm + (size_t)(k1 + rb)*DD + n0 + gb*4,     &Bs[nb][rb *PAD + gb *4]);
            WAIT_ASYNC(3);
        } else {
            WAIT_ASYNC(0);
        }
        __syncthreads();
        wmma_chunk_nn(As[cur], Bs[cur], wm, wn, l16, hi, c);
        __syncthreads();
    }
    // --- tail chunk: k0 = 992, 8 valid k, zero-padded, synchronous ---
    {
        float* AsT = As[1];
        float* BsT = Bs[1];
#pragma unroll
        for (int i = 0; i < 8; ++i) {
            int idx = t + i*256;
            int r = idx >> 5, kk = idx & 31;
            int mr = m0 + r;
            AsT[r*PAD + kk] = (kk < 8 && mr < NN) ? A[(size_t)mr * NN + 992 + kk] : 0.0f;
        }
#pragma unroll
        for (int i = 0; i < 4; ++i) {
            int idx = t + i*256;
            int rk = idx >> 5, nnx = idx & 31;
            BsT[rk*PAD + nnx] = (rk < 8) ? Bm[(size_t)(992 + rk) * DD + n0 + nnx] : 0.0f;
        }
        __syncthreads();
        wmma_chunk_nn(AsT, BsT, wm, wn, l16, hi, c);
    }
#else
    for (int k0 = 0; k0 < 1024; k0 += 32) {
#pragma unroll
        for (int i = 0; i < 8; ++i) {
            int idx = t + i*256;
            int r = idx >> 5, kk = idx & 31;
            int mr = m0 + r, kc = k0 + kk;
            As[0][r*PAD + kk] = (mr < NN && kc < NN) ? A[(size_t)mr * NN + kc] : 0.0f;
        }
#pragma unroll
        for (int i = 0; i < 4; ++i) {
            int idx = t + i*256;
            int r = idx >> 5, nnx = idx & 31;
            int kc = k0 + r;
            Bs[0][r*PAD + nnx] = (kc < NN) ? Bm[(size_t)kc * DD + n0 + nnx] : 0.0f;
        }
        __syncthreads();
        wmma_chunk_nn(As[0], Bs[0], wm, wn, l16, hi, c);
        __syncthreads();
    }
#endif

    int n = n0 + wn*16 + l16;
#pragma unroll
    for (int r = 0; r < 8; ++r) {
        int m = m0 + wm*16 + r + 8*hi;
        if (m < NN) C[(size_t)m * DD + n] = c[r];
    }
}

// ---------------------------------------------------------------------------
// Kernel 7: head GEMM: [att_feats | baf](16000x5120) * [cls_w;reg_w]^T(290x5120)
// ---------------------------------------------------------------------------
__global__ void gemm_head_wmma(const float* __restrict__ attf,
                               const float* __restrict__ baf,
                               const float* __restrict__ cls_w,
                               const float* __restrict__ cls_b,
                               const float* __restrict__ reg_w,
                               const float* __restrict__ reg_b,
                               float* __restrict__ C)
{
    __shared__ float As[2][64*PAD];
    __shared__ float Ws[2][32*PAD];
    const int t    = threadIdx.x;
    const int m0   = blockIdx.y * 64;
    const int n0   = blockIdx.x * 32;
    const int wave = t >> 5, lane = t & 31, l16 = lane & 15, hi = lane >> 4;
    const int wm   = wave >> 1, wn = wave & 1;

    v8f c = {0.f,0.f,0.f,0.f,0.f,0.f,0.f,0.f};

#ifdef USE_ASYNC
    const int ra0 = t >> 3,         ga0 = t & 7;
    const int ra1 = (t + 256) >> 3, ga1 = (t + 256) & 7;
    const int rw  = t >> 3,         gw  = t & 7;
    // per-lane weight row pointer (select, not branch: keep EXEC full and
    // exactly one async issue per thread per chunk)
    const int nw = n0 + rw;
    int nr = nw - NCC; nr = (nr < 0) ? 0 : ((nr >= 3*SS) ? 3*SS - 1 : nr);
    const float* wrow = (nw < NCC) ? (cls_w + (size_t)nw * DD2)
                                   : (reg_w + (size_t)nr * DD2);

    const int nch = DD2 / 32;                          // 160
    {
        const float* Ap = attf;                        // k0 = 0 < DD
        ASYNC_B128(Ap   + (size_t)(m0 + ra0)*DD + ga0*4, &As[0][ra0*PAD + ga0*4]);
        ASYNC_B128(Ap   + (size_t)(m0 + ra1)*DD + ga1*4, &As[0][ra1*PAD + ga1*4]);
        ASYNC_B128(wrow + gw*4,                          &Ws[0][rw *PAD + gw *4]);
    }
    for (int i = 0; i < nch; ++i) {
        const int cur = i & 1;
        if (i + 1 < nch) {
            const int k1 = (i + 1) * 32;
            const int nb = (i + 1) & 1;
            const float* Ap = (k1 < DD) ? attf : baf;  // concat along K (uniform)
            const int kb    = (k1 < DD) ? k1 : k1 - DD;
            ASYNC_B128(Ap   + (size_t)(m0 + ra0)*DD + kb + ga0*4, &As[nb][ra0*PAD + ga0*4]);
            ASYNC_B128(Ap   + (size_t)(m0 + ra1)*DD + kb + ga1*4, &As[nb][ra1*PAD + ga1*4]);
            ASYNC_B128(wrow + k1 + gw*4,                          &Ws[nb][rw *PAD + gw *4]);
            WAIT_ASYNC(3);
        } else {
            WAIT_ASYNC(0);
        }
        __syncthreads();
        wmma_chunk_nt(As[cur], Ws[cur], wm, wn, l16, hi, c);
        __syncthreads();
    }
#else
    for (int k0 = 0; k0 < DD2; k0 += 32) {
        const float* Ap = (k0 < DD) ? attf : baf;
        const int kb    = (k0 < DD) ? k0 : k0 - DD;
#pragma unroll
        for (int i = 0; i < 8; ++i) {
            int idx = t + i*256;
            int r = idx >> 5, kk = idx & 31;
            As[0][r*PAD + kk] = Ap[(size_t)(m0 + r) * DD + kb + kk];
        }
#pragma unroll
        for (int i = 0; i < 4; ++i) {
            int idx = t + i*256;
            int r = idx >> 5, kk = idx & 31;
            int n = n0 + r;
            float v = 0.0f;
            if (n < NCC)        v = cls_w[(size_t)n * DD2 + k0 + kk];
            else if (n < OUTH)  v = reg_w[(size_t)(n - NCC) * DD2 + k0 + kk];
            Ws[0][r*PAD + kk] = v;
        }
        __syncthreads();
        wmma_chunk_nt(As[0], Ws[0], wm, wn, l16, hi, c);
        __syncthreads();
    }
#endif

    int n = n0 + wn*16 + l16;
    if (n < OUTH) {
        float bv = (n < NCC) ? cls_b[n] : reg_b[n - NCC];
#pragma unroll
        for (int r = 0; r < 8; ++r) {
            int m = m0 + wm*16 + r + 8*hi;
            C[(size_t)m * OUTH + n] = c[r] + bv;
        }
    }
}

// ---------------------------------------------------------------------------
// Kernel 8: assemble reg_proposals (B,N,293)
// ---------------------------------------------------------------------------
__global__ void assemble_props(const float* __restrict__ head,
                               const float* __restrict__ anchors,
                               float* __restrict__ prop)
{
    int idx = blockIdx.x * blockDim.x + threadIdx.x;
    if (idx >= BB*NN*PROP) return;
    int cc = idx % PROP;
    int t  = idx / PROP;
    int n  = t % NN;
    const float* hrow = head + (size_t)t * OUTH;
    const float* arow = anchors + (size_t)n * PROP;
    float v;
    if (cc < NCC) {
        v = hrow[cc];
    } else if (cc < NCC + 3) {
        v = arow[cc];
    } else {
        int s = cc - (NCC + 3);
        float r = hrow[NCC + s];
        if (s >= 2*SS) r = 1.0f / (1.0f + __expf(-r));
        v = arow[cc] + r;
    }
    prop[idx] = v;
}

// ---------------------------------------------------------------------------
// Launcher
// ---------------------------------------------------------------------------
extern "C" void kernel_launch(void* const* d_in, const int* in_sizes, int n_in,
                              void* d_out, int out_size, void* d_ws, size_t ws_size,
                              hipStream_t stream)
{
    const float*         batch_features = (const float*)d_in[0];
    const float*         conv_w         = (const float*)d_in[1];
    const float*         conv_b         = (const float*)d_in[2];
    const int*           cut_xs         = (const int*)d_in[3];
    const unsigned char* invalid_mask   = (const unsigned char*)d_in[4];
    const float*         anchors        = (const float*)d_in[5];
    const float*         attn_w         = (const float*)d_in[6];
    const float*         attn_b         = (const float*)d_in[7];
    const float*         cls_w          = (const float*)d_in[8];
    const float*         cls_b          = (const float*)d_in[9];
    const float*         reg_w          = (const float*)d_in[10];
    const float*         reg_b          = (const float*)d_in[11];

    // d_out = reg_proposals | attn_mat | feats (flat, in return order)
    float* out   = (float*)d_out;
    float* prop  = out;
    float* attnm = out + (size_t)BB*NN*PROP;
    float* feats = attnm + (size_t)BB*NN*NN;

    // workspace
    float* ws     = (float*)d_ws;
    float* baf    = ws;
    float* scores = baf    + (size_t)BB*NN*DD;
    float* attf   = scores + (size_t)BB*NN*NM1;
    float* head   = attf   + (size_t)BB*NN*DD;

    conv1x1_wmma<<<3840, 256, 0, stream>>>(batch_features, conv_w, conv_b, feats);

    gather_baf<<<(BB*NN*HH + 255)/256, 256, 0, stream>>>(feats, cut_xs, invalid_mask, baf);

    {
        dim3 g((NM1 + 31)/32, (BB*NN)/64);
        gemm_nt_wmma<<<g, 256, 0, stream>>>(baf, attn_w, attn_b, scores,
                                            BB*NN, NM1, DD);
    }

    softmax_rows<<<BB*NN, 256, 0, stream>>>(scores);

    build_attn<<<(int)(((size_t)BB*NN*NN + 255)/256), 256, 0, stream>>>(scores, attnm);

    {
        dim3 g(DD/32, (NN + 63)/64, BB);
        gemm_nn_wmma<<<g, 256, 0, stream>>>(attnm, baf, attf);
    }

    {
        dim3 g((OUTH + 31)/32, (BB*NN)/64);
        gemm_head_wmma<<<g, 256, 0, stream>>>(attf, baf, cls_w, cls_b, reg_w, reg_b, head);
    }

    assemble_props<<<(BB*NN*PROP + 255)/256, 256, 0, stream>>>(head, anchors, prop);
}